// MinibatchDiscrimination_4587025072595
// MI455X (gfx1250) — compile-verified
//
#include <hip/hip_runtime.h>
#include <math.h>

typedef __attribute__((ext_vector_type(2))) float v2f;
typedef __attribute__((ext_vector_type(4))) float v4f;
typedef __attribute__((ext_vector_type(8))) float v8f;

#define BSZ   512
#define INF   512
#define OUTF  64
#define KD    16
#define NCOL  (OUTF * KD)   // 1024

// ---------------------------------------------------------------------------
// Phase 1: matrices = x @ T   (512x512 @ 512x1024), stored transposed as
// M[o][b][k]  (OUT_F, B, KD) so phase 2 reads one contiguous 32KB slice per o.
// One wave per 16x16 output tile; K-loop of v_wmma_f32_16x16x4_f32.
// ---------------------------------------------------------------------------
__global__ __launch_bounds__(32) void mbdisc_gemm_wmma(
    const float* __restrict__ x,   // (512, 512) row-major
    const float* __restrict__ T,   // (512, 1024) row-major
    float* __restrict__ M)         // (64, 512, 16)
{
    const int mt   = blockIdx.x;          // 0..31  : batch tile
    const int o    = blockIdx.y;          // 0..63  : output feature == N tile
    const int lane = threadIdx.x;         // 0..31 (wave32)
    const int half = lane >> 4;           // 0 or 1
    const int lid  = lane & 15;

    const int m0 = mt * 16;
    const int n0 = o * 16;

    // A: lane holds x[m0+lid][k + 2*half], x[m0+lid][k + 2*half + 1]
    const float* xrow = x + (size_t)(m0 + lid) * INF + 2 * half;
    // B: lane holds T[k + 2*half][n0+lid], T[k + 2*half + 1][n0+lid]
    const float* tcol = T + (size_t)(2 * half) * NCOL + n0 + lid;

    v8f acc = {};

    #pragma unroll 4
    for (int k = 0; k < INF; k += 4) {
        // contiguous, 8B-aligned (k multiple of 4, 2*half even)
        v2f a = *(const v2f*)(xrow + k);
        v2f b;
        b.x = tcol[(size_t)k * NCOL];
        b.y = tcol[(size_t)k * NCOL + NCOL];
        // 8 args: (neg_a, A, neg_b, B, c_mod, C, reuse_a, reuse_b)
        acc = __builtin_amdgcn_wmma_f32_16x16x4_f32(
            false, a, false, b, (short)0, acc, false, false);
    }

    // D layout: VGPR r, lanes 0-15 -> D[r][lid]; lanes 16-31 -> D[r+8][lid]
    // Column n0+lid of the 1024-wide result = (o, k=lid); row = batch index.
    float* Mo = M + (size_t)o * BSZ * KD;
    #pragma unroll
    for (int r = 0; r < 8; ++r) {
        const int m = m0 + r + half * 8;   // batch index
        Mo[(size_t)m * KD + lid] = acc[r];
    }
}

// ---------------------------------------------------------------------------
// Phase 2: out[j][o] = w[j] * sum_i exp(-sum_k |M[i,o,k]-M[j,o,k]|) - w[j]
// One block per o (64 blocks), 512 threads = one j per thread.
// The 512x16 fp32 slice M[o,:,:] (32KB) is staged in LDS; the inner i-loop is
// an LDS-broadcast read shared by all lanes.
// ---------------------------------------------------------------------------
__global__ __launch_bounds__(512) void mbdisc_pairwise(
    const float* __restrict__ M,   // (64, 512, 16)
    const float* __restrict__ w,   // (512,)
    float* __restrict__ out)       // (512, 64)
{
    __shared__ v4f s[BSZ * 4];     // 512 rows x 16 floats = 32 KB

    const int o = blockIdx.x;
    const int j = threadIdx.x;

    const v4f* Mo = (const v4f*)(M + (size_t)o * BSZ * KD);
    #pragma unroll
    for (int t = 0; t < 4; ++t)
        s[j + t * BSZ] = Mo[j + t * BSZ];
    __syncthreads();

    const v4f a0 = s[j * 4 + 0];
    const v4f a1 = s[j * 4 + 1];
    const v4f a2 = s[j * 4 + 2];
    const v4f a3 = s[j * 4 + 3];

    float acc = 0.0f;
    #pragma unroll 2
    for (int i = 0; i < BSZ; ++i) {
        const v4f b0 = s[i * 4 + 0];
        const v4f b1 = s[i * 4 + 1];
        const v4f b2 = s[i * 4 + 2];
        const v4f b3 = s[i * 4 + 3];
        float n;
        n  = fabsf(a0.x - b0.x); n += fabsf(a0.y - b0.y);
        n += fabsf(a0.z - b0.z); n += fabsf(a0.w - b0.w);
        n += fabsf(a1.x - b1.x); n += fabsf(a1.y - b1.y);
        n += fabsf(a1.z - b1.z); n += fabsf(a1.w - b1.w);
        n += fabsf(a2.x - b2.x); n += fabsf(a2.y - b2.y);
        n += fabsf(a2.z - b2.z); n += fabsf(a2.w - b2.w);
        n += fabsf(a3.x - b3.x); n += fabsf(a3.y - b3.y);
        n += fabsf(a3.z - b3.z); n += fabsf(a3.w - b3.w);
        acc += __expf(-n);
    }

    const float wj = w[j];
    out[(size_t)j * OUTF + o] = wj * acc - wj;
}

extern "C" void kernel_launch(void* const* d_in, const int* in_sizes, int n_in,
                              void* d_out, int out_size, void* d_ws, size_t ws_size,
                              hipStream_t stream) {
    const float* x = (const float*)d_in[0];   // (512, 512)
    const float* w = (const float*)d_in[1];   // (1, 512)
    const float* T = (const float*)d_in[2];   // (512, 64, 16) == (512, 1024)
    float* M   = (float*)d_ws;                // needs 64*512*16*4 = 2 MB
    float* out = (float*)d_out;               // (512, 64)

    dim3 g1(BSZ / 16, OUTF);                  // 32 x 64 tiles, 1 wave each
    mbdisc_gemm_wmma<<<g1, 32, 0, stream>>>(x, T, M);

    mbdisc_pairwise<<<OUTF, BSZ, 0, stream>>>(M, w, out);
}